// Feature_Extract_73658689126815
// MI455X (gfx1250) — compile-verified
//
#include <hip/hip_runtime.h>
#include <cstdint>

#ifndef __has_builtin
#define __has_builtin(x) 0
#endif

#define D_FEAT 64
#define OUT_STRIDE 192

// ---------------- CDNA5 async-to-LDS staging (gfx1250) -----------------
#if defined(__gfx1250__) && __has_builtin(__builtin_amdgcn_global_load_async_to_lds_b128)
#define USE_ASYNC 1
typedef int v4i_async __attribute__((vector_size(16)));
typedef __attribute__((address_space(1))) v4i_async* gptr_v4i;
typedef __attribute__((address_space(3))) v4i_async* lptr_v4i;
__device__ __forceinline__ void async_ld_b128(const float* g, float* l) {
    // GLOBAL_LOAD_ASYNC_TO_LDS_B128: per-lane 16B global -> LDS, tracked by ASYNCcnt
    __builtin_amdgcn_global_load_async_to_lds_b128(
        (gptr_v4i)(uintptr_t)g,
        (lptr_v4i)(uint32_t)(uintptr_t)l,   // low 32 bits of generic LDS addr = LDS offset
        0, 0);
}
#if __has_builtin(__builtin_amdgcn_s_wait_asynccnt)
#define ASYNC_WAIT(n) __builtin_amdgcn_s_wait_asynccnt(n)
#else
#define ASYNC_WAIT(n) asm volatile("s_wait_asynccnt %0" :: "i"(n) : "memory")
#endif
#else
#define USE_ASYNC 0
#endif

// ---------------- K1: deg = 1.0 (self-loop weight) -----------------
__global__ void k_deg_init(float* __restrict__ deg, int n) {
    int i = blockIdx.x * blockDim.x + threadIdx.x;
    if (i < n) deg[i] = 1.0f;
}

// ---------------- K2: deg[row[e]] += w[e] -----------------
__global__ void k_deg_acc(const int* __restrict__ row, const float* __restrict__ w,
                          float* __restrict__ deg, int e) {
    int stride = gridDim.x * blockDim.x;
    for (int i = blockIdx.x * blockDim.x + threadIdx.x; i < e; i += stride) {
        if (i + stride < e) {
            __builtin_prefetch(&row[i + stride], 0, 0);  // global_prefetch_b8
            __builtin_prefetch(&w[i + stride], 0, 0);
        }
        atomicAdd(deg + row[i], w[i]);
    }
}

// ---------------- K3: in-place dis = deg > 0 ? rsqrt(deg) : 0 -----------------
__global__ void k_dis(float* __restrict__ deg, int n) {
    int i = blockIdx.x * blockDim.x + threadIdx.x;
    if (i < n) {
        float d = deg[i];
        deg[i] = (d > 0.0f) ? __frsqrt_rn(d) : 0.0f;
    }
}

// ---------------- K4: out[:,0:64] = x ; out[:,64:128] = dis^2 * x (x1 self-loop init)
__global__ void k_self1(const float* __restrict__ x, const float* __restrict__ dis,
                        const int* __restrict__ hfp, float* __restrict__ out, int n) {
    int t = blockIdx.x * blockDim.x + threadIdx.x;
    int node = t >> 4, q = t & 15;
    if (node >= n) return;
    const float4 xv = *(const float4*)(x + (size_t)node * D_FEAT + q * 4);
    float s = 0.0f;
    if (hfp[0]) { float di = dis[node]; s = di * di; }
    float* o = out + (size_t)node * OUT_STRIDE + q * 4;
    *(float4*)o = xv;
    float4 m; m.x = s * xv.x; m.y = s * xv.y; m.z = s * xv.z; m.w = s * xv.w;
    *(float4*)(o + D_FEAT) = m;
}

// ---------------- K6: out[:,128:192] = dis^2 * x1   (x2 self-loop init; x1 complete)
__global__ void k_self2(const float* __restrict__ dis, const int* __restrict__ hfp,
                        float* __restrict__ out, int n) {
    int t = blockIdx.x * blockDim.x + threadIdx.x;
    int node = t >> 4, q = t & 15;
    if (node >= n) return;
    const float4 x1 = *(const float4*)(out + (size_t)node * OUT_STRIDE + D_FEAT + q * 4);
    float s = 0.0f;
    if (hfp[0]) { float di = dis[node]; s = di * di; }
    float4 m; m.x = s * x1.x; m.y = s * x1.y; m.z = s * x1.z; m.w = s * x1.w;
    *(float4*)(out + (size_t)node * OUT_STRIDE + 2 * D_FEAT + q * 4) = m;
}

// ---------------- K5/K7: edge propagate  dst[col] += norm * src[row] ----------------
// 16 lanes per edge (float4 each). 4-deep async-to-LDS pipeline for the random 256B
// src-row gather, with edge metadata (row/col/w) software-pipelined one stage ahead
// of the async issue so neither latency chain serializes the loop.
#define PROP_BLOCK 256
#define PROP_GRID  8192
#define DEPTH 4
__global__ void k_prop(const int* __restrict__ row, const int* __restrict__ col,
                       const float* __restrict__ w, const float* __restrict__ dis,
                       const int* __restrict__ hfp,
                       const float* __restrict__ src, int sstride,
                       float* __restrict__ dst, int dstride, int e) {
#if USE_ASYNC
    __shared__ float stage[(PROP_BLOCK / 16) * DEPTH * D_FEAT];  // 16 KB
    float* slots = &stage[(threadIdx.x >> 4) * (DEPTH * D_FEAT)];
#endif
    const int tid = blockIdx.x * blockDim.x + threadIdx.x;
    const int lane = tid & 15;
    const int g0 = tid >> 4;
    const int ngroups = (gridDim.x * blockDim.x) >> 4;
    const int hf = hfp[0];
    const int iters = (e + ngroups - 1) / ngroups;  // uniform: no early return, waits are wave-wide

    // metadata pipeline registers: edges it+0 .. it+3 (clamped to (0,0,0) when out of range)
    int r0 = 0, c0 = 0; float w0 = 0.0f;
    int r1 = 0, c1 = 0; float w1 = 0.0f;
    int r2 = 0, c2 = 0; float w2 = 0.0f;
    int r3 = 0, c3 = 0; float w3 = 0.0f;
    {
        int g;
        g = g0;               if (g < e) { r0 = row[g]; c0 = col[g]; w0 = w[g]; }
        g = g0 + ngroups;     if (g < e) { r1 = row[g]; c1 = col[g]; w1 = w[g]; }
        g = g0 + 2 * ngroups; if (g < e) { r2 = row[g]; c2 = col[g]; w2 = w[g]; }
        g = g0 + 3 * ngroups; if (g < e) { r3 = row[g]; c3 = col[g]; w3 = w[g]; }
    }
#if USE_ASYNC
    // prologue: issue async gathers for edges it=0,1,2 into slots 0,1,2
    async_ld_b128(src + (size_t)r0 * sstride + lane * 4, slots + 0 * D_FEAT + lane * 4);
    async_ld_b128(src + (size_t)r1 * sstride + lane * 4, slots + 1 * D_FEAT + lane * 4);
    async_ld_b128(src + (size_t)r2 * sstride + lane * 4, slots + 2 * D_FEAT + lane * 4);
#endif

    for (int it = 0; it < iters; ++it) {
#if USE_ASYNC
        // issue async for edge it+3 (metadata already resident in r3)
        async_ld_b128(src + (size_t)r3 * sstride + lane * 4,
                      slots + ((it + 3) & (DEPTH - 1)) * D_FEAT + lane * 4);
#endif
        // load metadata for edge it+4; prefetch it+5's metadata cachelines
        const int gn = g0 + (it + 4) * ngroups;
        int nr = 0, nc = 0; float nw = 0.0f;
        if (gn < e) {
            nr = row[gn]; nc = col[gn]; nw = w[gn];
            const int gp = gn + ngroups;
            if (gp < e) {
                __builtin_prefetch(&row[gp], 0, 0);   // global_prefetch_b8
                __builtin_prefetch(&col[gp], 0, 0);
                __builtin_prefetch(&w[gp], 0, 0);
            }
        }
#if USE_ASYNC
        ASYNC_WAIT(3);  // 3 younger gathers may stay in flight; edge `it`'s slot is ready
#else
        __builtin_prefetch(src + (size_t)r3 * sstride + lane * 4, 0, 0);
#endif
        // consume edge it
        const int gc = g0 + it * ngroups;
        if (gc < e) {
            const float nrm = hf ? (dis[r0] * w0 * dis[c0]) : w0;
            float4 xv;
#if USE_ASYNC
            xv = *(const float4*)(slots + (it & (DEPTH - 1)) * D_FEAT + lane * 4);
#else
            xv = *(const float4*)(src + (size_t)r0 * sstride + lane * 4);
#endif
            float* d = dst + (size_t)c0 * dstride + lane * 4;
            atomicAdd(d + 0, nrm * xv.x);
            atomicAdd(d + 1, nrm * xv.y);
            atomicAdd(d + 2, nrm * xv.z);
            atomicAdd(d + 3, nrm * xv.w);
        }
        // shift metadata pipeline
        r0 = r1; c0 = c1; w0 = w1;
        r1 = r2; c1 = c2; w1 = w2;
        r2 = r3; c2 = c3; w2 = w3;
        r3 = nr; c3 = nc; w3 = nw;
    }
}

extern "C" void kernel_launch(void* const* d_in, const int* in_sizes, int n_in,
                              void* d_out, int out_size, void* d_ws, size_t ws_size,
                              hipStream_t stream) {
    const float* x  = (const float*)d_in[0];
    const int*   ei = (const int*)d_in[1];     // [2, E]
    const float* w  = (const float*)d_in[2];
    const int*   hf = (const int*)d_in[3];     // scalar

    const int n = in_sizes[0] / D_FEAT;
    const int e = in_sizes[2];
    const int* row = ei;
    const int* col = ei + e;

    float* out = (float*)d_out;                // [n, 192]
    float* deg = (float*)d_ws;                 // n floats; becomes dis in-place

    const int nb_n  = (n + 255) / 256;
    const int nb_nf = (n * 16 + 255) / 256;    // node x float4 threads

    // degree (self-loop folded into init), then dis = rsqrt(deg) in place
    k_deg_init<<<nb_n, 256, 0, stream>>>(deg, n);
    k_deg_acc<<<2048, 256, 0, stream>>>(row, w, deg, e);
    k_dis<<<nb_n, 256, 0, stream>>>(deg, n);

    // out[:,0:64] = x ; out[:,64:128] = dis^2 * x
    k_self1<<<nb_nf, 256, 0, stream>>>(x, deg, hf, out, n);
    // x1 edges: out[:,64:128] += norm * x[row]
    k_prop<<<PROP_GRID, PROP_BLOCK, 0, stream>>>(row, col, w, deg, hf,
                                                 x, D_FEAT,
                                                 out + D_FEAT, OUT_STRIDE, e);
    // out[:,128:192] = dis^2 * x1
    k_self2<<<nb_nf, 256, 0, stream>>>(deg, hf, out, n);
    // x2 edges: out[:,128:192] += norm * x1[row]
    k_prop<<<PROP_GRID, PROP_BLOCK, 0, stream>>>(row, col, w, deg, hf,
                                                 out + D_FEAT, OUT_STRIDE,
                                                 out + 2 * D_FEAT, OUT_STRIDE, e);
}